// ToSparse_2748779070168
// MI455X (gfx1250) — compile-verified
//
#include <hip/hip_runtime.h>
#include <hip/hip_bf16.h>

typedef float v2f __attribute__((ext_vector_type(2)));
typedef float v8f __attribute__((ext_vector_type(8)));

#define CHUNK 4096            // elements compacted per (single-wave) block
#define GROUP 128             // 32 lanes * 4 elements (b128 loads) per iteration
#define ITERS (CHUNK / GROUP) // 32

// ---------------------------------------------------------------------------
// Phase 0: zero the whole output (indices / mod / values tails must be 0).
// ---------------------------------------------------------------------------
__global__ void zero_out_kernel(float4* __restrict__ p4, int n4,
                                float* __restrict__ tail, int ntail) {
  int i = blockIdx.x * blockDim.x + threadIdx.x;
  int stride = gridDim.x * blockDim.x;
  float4 z; z.x = 0.f; z.y = 0.f; z.z = 0.f; z.w = 0.f;
  for (; i < n4; i += stride) p4[i] = z;
  if (blockIdx.x == 0 && threadIdx.x < ntail) tail[threadIdx.x] = 0.f;
}

// Zero the padding tail of the counts array so the WMMA scan can load
// full 256-count tiles unguarded (keeps EXEC all-ones through the WMMAs).
__global__ void pad_counts_kernel(int* __restrict__ counts, int NB, int NBpad) {
  int i = NB + blockIdx.x * blockDim.x + threadIdx.x;
  if (i < NBpad) counts[i] = 0;
}

// ---------------------------------------------------------------------------
// Phase 1: per-block nonzero counts (one wave32 per CHUNK, b128 loads).
// ---------------------------------------------------------------------------
__global__ __launch_bounds__(32) void count_kernel(const float* __restrict__ x,
                                                   int* __restrict__ counts,
                                                   int total) {
  const int lane = threadIdx.x;
  const long long base = (long long)blockIdx.x * CHUNK;
  const float4* __restrict__ x4 = (const float4*)x;
  int cnt = 0;
  for (int it = 0; it < ITERS; ++it) {
    long long g4 = (base >> 2) + it * 32 + lane;  // float4 index
    __builtin_prefetch(&x4[g4 + 64], 0, 1);       // global_prefetch_b8
    float4 v = x4[g4];
    cnt += (v.x != 0.f) + (v.y != 0.f) + (v.z != 0.f) + (v.w != 0.f);
  }
  // wave32 reduction
  for (int off = 16; off > 0; off >>= 1) cnt += __shfl_down(cnt, off, 32);
  if (lane == 0) counts[blockIdx.x] = cnt;
}

// ---------------------------------------------------------------------------
// Phase 2: exclusive scan of block counts using v_wmma_f32_16x16x4_f32.
// One wave; each 256-count tile is laid out as a 16x16 matrix X (row-major),
// and D = X * T (T upper-triangular ones) gives inclusive row prefixes.
// K=16 product is built from 4 chained K=4 WMMAs. Cross-row offsets + carry
// are fixed up serially (16 values). All counts < 2^24 -> f32 exact.
// counts[] is padded with zeros to a 256-multiple, so fragment loads are
// unguarded b64 loads and EXEC stays all-ones through the WMMA chain.
//
// Assumed VGPR layouts (CDNA5 ISA, 16x16x4 f32):
//   A 16x4 : lane L -> M = L&15 ; VGPR0 = K = (L>>4)*2, VGPR1 = K+1
//   B 4x16 : lane L -> N = L&15 ; VGPR0 = K = (L>>4)*2, VGPR1 = K+1
//   C 16x16: lane L, VGPR r -> M = r + (L>>4)*8, N = L&15
// ---------------------------------------------------------------------------
__global__ __launch_bounds__(32) void scan_kernel(const int* __restrict__ counts,
                                                  int* __restrict__ offsets,
                                                  float* __restrict__ nnz_out,
                                                  int NB) {
  const int lane = threadIdx.x;
  const int MN   = lane & 15;        // M for A / N for B,C
  const int half = lane >> 4;        // 0 or 1
  const int kHi  = half * 2;
  const int nTiles = (NB + 255) >> 8;
  const int2* __restrict__ counts2 = (const int2*)counts;

  float carry = 0.f;
  for (int t = 0; t < nTiles; ++t) {
    const int base = t << 8;
    v8f acc = {};
    #pragma unroll
    for (int j = 0; j < 4; ++j) {
      const int k0 = 4 * j;
      // A fragment: X[M][k0+kHi], X[M][k0+kHi+1]  (paired -> global_load_b64)
      const int2 cv = counts2[(base + 16 * MN + k0 + kHi) >> 1];
      v2f a;
      a.x = (float)cv.x;
      a.y = (float)cv.y;
      // B fragment: T[k][n] = (k <= n)
      v2f b;
      b.x = ((k0 + kHi)     <= MN) ? 1.f : 0.f;
      b.y = ((k0 + kHi + 1) <= MN) ? 1.f : 0.f;
      acc = __builtin_amdgcn_wmma_f32_16x16x4_f32(
          /*neg_a=*/false, a, /*neg_b=*/false, b,
          /*c_mod=*/(short)0, acc, /*reuse_a=*/false, /*reuse_b=*/false);
    }
    // Row totals live at N==15 -> lanes 15 (M=r) and 31 (M=r+8).
    float rowTot[16];
    #pragma unroll
    for (int r = 0; r < 8; ++r) {
      rowTot[r]     = __shfl(acc[r], 15, 32);
      rowTot[r + 8] = __shfl(acc[r], 31, 32);
    }
    float R[16];
    float s = 0.f;
    #pragma unroll
    for (int m = 0; m < 16; ++m) { R[m] = s; s += rowTot[m]; }
    // Emit exclusive offsets: excl = carry + R[m] + incl(m,n) - X[m,n]
    #pragma unroll
    for (int r = 0; r < 8; ++r) {
      const int m   = r + half * 8;
      const int idx = base + 16 * m + MN;
      if (idx < NB) {
        const float xv = (float)counts[idx];
        offsets[idx] = (int)(carry + R[m] + acc[r] - xv);
      }
    }
    carry += s;
  }
  if (lane == 0) nnz_out[0] = carry;   // nnz as exact float (< 2^24)
}

// ---------------------------------------------------------------------------
// Phase 3: stable scatter. One wave per CHUNK; within each 128-element group
// ranks come from 4 wave32 ballots (lane-major element order preserved).
// ---------------------------------------------------------------------------
__global__ __launch_bounds__(32) void scatter_kernel(const float* __restrict__ x,
                                                     const int* __restrict__ offsets,
                                                     const int* __restrict__ n_units,
                                                     float* __restrict__ idx_out,
                                                     float* __restrict__ mod_out,
                                                     float* __restrict__ val_out,
                                                     int total) {
  const int lane = threadIdx.x;
  const int N = *n_units;
  const long long base = (long long)blockIdx.x * CHUNK;
  const float4* __restrict__ x4 = (const float4*)x;
  int running = offsets[blockIdx.x];
  const unsigned below = (1u << lane) - 1u;

  for (int it = 0; it < ITERS; ++it) {
    const long long g4 = (base >> 2) + it * 32 + lane;
    __builtin_prefetch(&x4[g4 + 64], 0, 1);
    const float4 v = x4[g4];
    const bool nz0 = (v.x != 0.f), nz1 = (v.y != 0.f),
               nz2 = (v.z != 0.f), nz3 = (v.w != 0.f);
    const unsigned m0 = __builtin_amdgcn_ballot_w32(nz0);
    const unsigned m1 = __builtin_amdgcn_ballot_w32(nz1);
    const unsigned m2 = __builtin_amdgcn_ballot_w32(nz2);
    const unsigned m3 = __builtin_amdgcn_ballot_w32(nz3);
    int p = running + __popc(m0 & below) + __popc(m1 & below) +
                      __popc(m2 & below) + __popc(m3 & below);
    const int gi = (int)(base + it * GROUP) + lane * 4;

    #pragma unroll
    for (int j = 0; j < 4; ++j) {
      const bool  nz = (j == 0) ? nz0 : (j == 1) ? nz1 : (j == 2) ? nz2 : nz3;
      const float vv = (j == 0) ? v.x : (j == 1) ? v.y : (j == 2) ? v.z : v.w;
      if (nz) {
        const int g   = gi + j;
        const int row = g / N;
        const int col = g - row * N;
        idx_out[2 * p]     = (float)row;
        idx_out[2 * p + 1] = (float)col;
        mod_out[p]         = (float)col;   // order % n_units
        val_out[p]         = vv;
        ++p;
      }
    }
    running += __popc(m0) + __popc(m1) + __popc(m2) + __popc(m3);
  }
}

// ---------------------------------------------------------------------------
extern "C" void kernel_launch(void* const* d_in, const int* in_sizes, int n_in,
                              void* d_out, int out_size, void* d_ws, size_t ws_size,
                              hipStream_t stream) {
  const float* x       = (const float*)d_in[0];
  const int*   n_units = (const int*)d_in[1];
  const int total = in_sizes[0];                    // B*N = 16,777,216
  const int NB    = (total + CHUNK - 1) / CHUNK;    // 4096
  const int NBpad = ((NB + 255) >> 8) << 8;         // tile multiple of 256

  float* out = (float*)d_out;
  float* idx_out = out;                              // [total, 2]
  float* mod_out = out + 2 * (size_t)total;          // [total]
  float* val_out = out + 3 * (size_t)total;          // [total]
  float* nnz_out = out + (size_t)(out_size - 1);     // scalar

  int* counts  = (int*)d_ws;           // NBpad ints (padded with zeros)
  int* offsets = counts + NBpad;       // NB ints

  // Phase 0: zero all outputs (tails must be zero; harness never re-poisons).
  const int n4    = out_size >> 2;
  const int ntail = out_size - (n4 << 2);
  zero_out_kernel<<<2048, 256, 0, stream>>>((float4*)out, n4,
                                            out + (size_t)n4 * 4, ntail);
  // Phase 1: per-block counts (+ zero the tile padding of counts[]).
  count_kernel<<<NB, 32, 0, stream>>>(x, counts, total);
  pad_counts_kernel<<<1, 256, 0, stream>>>(counts, NB, NBpad);
  // Phase 2: WMMA-based exclusive scan of block counts + nnz.
  scan_kernel<<<1, 32, 0, stream>>>(counts, offsets, nnz_out, NB);
  // Phase 3: stable compaction scatter.
  scatter_kernel<<<NB, 32, 0, stream>>>(x, offsets, n_units,
                                        idx_out, mod_out, val_out, total);
}